// SlotAttention_48808008352242
// MI455X (gfx1250) — compile-verified
//
#include <hip/hip_runtime.h>
#include <math.h>
#include <stdint.h>

#define DIM    192
#define HIDN   128
#define NSLOT  11
#define KPAD   16
#define BATCH  64
#define NTOK   4096
#define NITERS 3
#define EPS_   1e-8f
#define LN_EPS_ 1e-5f
#define SCALE_ 0.07216878364870323f   // 192^-0.5

typedef float v2f __attribute__((ext_vector_type(2)));
typedef float v8f __attribute__((ext_vector_type(8)));

__device__ __forceinline__ v8f wmma4(v2f a, v2f b, v8f c) {
  // V_WMMA_F32_16X16X4_F32 : D = A(16x4) * B(4x16) + C(16x16)
  return __builtin_amdgcn_wmma_f32_16x16x4_f32(
      false, a, false, b, (short)0, c, false, false);
}

__device__ __forceinline__ v8f vzero8() {
  v8f z;
#pragma unroll
  for (int e = 0; e < 8; ++e) z[e] = 0.f;
  return z;
}

// --- async global->LDS staging helpers (gfx1250 GLOBAL_LOAD_ASYNC_TO_LDS) ---
#if __has_builtin(__builtin_amdgcn_global_load_async_to_lds_b128)
#define HAS_ASYNC_LDS 1
typedef int v4i_vs __attribute__((vector_size(16)));
typedef __attribute__((address_space(3))) v4i_vs* lds_v4p;
typedef __attribute__((address_space(1))) v4i_vs* glb_v4p;
__device__ __forceinline__ void async_copy_b128(const float* g, float* l) {
  // generic LDS address truncates to the LDS byte offset (flat addressing rule)
  __builtin_amdgcn_global_load_async_to_lds_b128(
      (glb_v4p)(uintptr_t)g, (lds_v4p)(uint32_t)(uintptr_t)l, 0, 0);
}
__device__ __forceinline__ void wait_async0() {
#if __has_builtin(__builtin_amdgcn_s_wait_asynccnt)
  __builtin_amdgcn_s_wait_asynccnt(0);
#else
  asm volatile("s_wait_asynccnt 0x0" ::: "memory");
#endif
}
#endif

// ---------------------------------------------------------------------------
// Kernel A: LayerNorm(inputs) then key = x@Wk+bk, val = x@Wv+bv
// grid: (B*N)/64 blocks, 256 threads (8 waves). 64 rows per block in LDS,
// staged with async global->LDS b128 copies. 24 col-tile jobs (12 key +
// 12 val) -> 3 per wave; each wave does 4 M-tiles so the B fragment is
// reused 4x per k-step.
// ---------------------------------------------------------------------------
__global__ __launch_bounds__(256) void ln_kv_kernel(
    const float* __restrict__ inp, const float* __restrict__ lnw,
    const float* __restrict__ lnb,
    const float* __restrict__ Wk, const float* __restrict__ bk,
    const float* __restrict__ Wv, const float* __restrict__ bv,
    float* __restrict__ key, float* __restrict__ val) {
  __shared__ __align__(16) float xs[64 * DIM];
  __shared__ float redA[256], redB[256], mean_s[64], inv_s[64];
  const int tid = threadIdx.x;
  const size_t row0 = (size_t)blockIdx.x * 64;

#ifdef HAS_ASYNC_LDS
  for (int i = tid * 4; i < 64 * DIM; i += 256 * 4)
    async_copy_b128(inp + row0 * DIM + i, &xs[i]);
  wait_async0();
#else
  for (int i = tid; i < 64 * DIM; i += 256) xs[i] = inp[row0 * DIM + i];
#endif
  __syncthreads();

  { // per-row mean/var, 4 partial lanes per row
    int r = tid >> 2, p = tid & 3;
    float s = 0.f, s2 = 0.f;
    for (int k = p; k < DIM; k += 4) { float v = xs[r * DIM + k]; s += v; s2 += v * v; }
    redA[tid] = s; redB[tid] = s2;
  }
  __syncthreads();
  if ((tid & 3) == 0) {
    int r = tid >> 2;
    float s  = redA[tid] + redA[tid + 1] + redA[tid + 2] + redA[tid + 3];
    float s2 = redB[tid] + redB[tid + 1] + redB[tid + 2] + redB[tid + 3];
    float mu = s * (1.f / DIM);
    float var = s2 * (1.f / DIM) - mu * mu;
    mean_s[r] = mu; inv_s[r] = rsqrtf(var + LN_EPS_);
  }
  __syncthreads();
  for (int i = tid; i < 64 * DIM; i += 256) {
    int r = i / DIM, k = i - r * DIM;
    xs[i] = (xs[i] - mean_s[r]) * inv_s[r] * lnw[k] + lnb[k];
  }
  __syncthreads();

  const int wave = tid >> 5, lane = tid & 31;
  const int lm = lane & 15, lh = lane >> 4;
  for (int jj = 0; jj < 3; ++jj) {
    int job = wave * 3 + jj;                 // 0..23
    const float* W    = (job < 12) ? Wk : Wv;
    const float* bias = (job < 12) ? bk : bv;
    float*       dst  = (job < 12) ? key : val;
    int n0 = (job % 12) * 16;

    v8f acc[4];
#pragma unroll
    for (int mt = 0; mt < 4; ++mt) acc[mt] = vzero8();

    for (int kk = 0; kk < DIM / 4; ++kk) {
      int k0 = kk * 4 + lh * 2;
      v2f bf;
      bf.x = W[(size_t)k0 * DIM + n0 + lm];
      bf.y = W[(size_t)(k0 + 1) * DIM + n0 + lm];
#pragma unroll
      for (int mt = 0; mt < 4; ++mt) {
        v2f af;
        af.x = xs[(mt * 16 + lm) * DIM + k0];
        af.y = xs[(mt * 16 + lm) * DIM + k0 + 1];
        acc[mt] = wmma4(af, bf, acc[mt]);
      }
    }
    float bb = bias[n0 + lm];
#pragma unroll
    for (int mt = 0; mt < 4; ++mt)
#pragma unroll
      for (int v = 0; v < 8; ++v) {
        size_t row = row0 + mt * 16 + v + lh * 8;
        dst[row * DIM + n0 + lm] = acc[mt][v] + bb;
      }
  }
}

// ---------------------------------------------------------------------------
// slots = mu + sigma * noise
// ---------------------------------------------------------------------------
__global__ void init_slots_kernel(const float* __restrict__ noise,
                                  const float* __restrict__ mu,
                                  const float* __restrict__ sg,
                                  float* __restrict__ slots) {
  int i = blockIdx.x * 256 + threadIdx.x;
  if (i < BATCH * NSLOT * DIM) {
    int d = i % DIM;
    slots[i] = mu[d] + sg[d] * noise[i];
  }
}

// ---------------------------------------------------------------------------
// q = LN(slots) @ Wq + bq, stored padded to 16 rows per batch.
// grid: BATCH blocks, 128 threads (4 waves), 3 col-tiles per wave.
// ---------------------------------------------------------------------------
__global__ __launch_bounds__(128) void slots_q_kernel(
    const float* __restrict__ slots, const float* __restrict__ lnw,
    const float* __restrict__ lnb, const float* __restrict__ Wq,
    const float* __restrict__ bq, float* __restrict__ qpad) {
  __shared__ __align__(16) float sn[KPAD * DIM];
  const int b = blockIdx.x, tid = threadIdx.x;
  for (int i = tid; i < KPAD * DIM; i += 128)
    sn[i] = (i < NSLOT * DIM) ? slots[(size_t)b * NSLOT * DIM + i] : 0.f;
  __syncthreads();
  if (tid < NSLOT) {
    float s = 0.f, s2 = 0.f;
    for (int k = 0; k < DIM; ++k) { float v = sn[tid * DIM + k]; s += v; s2 += v * v; }
    float mu = s * (1.f / DIM);
    float inv = rsqrtf(s2 * (1.f / DIM) - mu * mu + LN_EPS_);
    for (int k = 0; k < DIM; ++k)
      sn[tid * DIM + k] = (sn[tid * DIM + k] - mu) * inv * lnw[k] + lnb[k];
  }
  __syncthreads();

  const int wave = tid >> 5, lane = tid & 31;
  const int lm = lane & 15, lh = lane >> 4;
  for (int jj = 0; jj < 3; ++jj) {
    int n0 = (wave * 3 + jj) * 16;
    v8f acc = vzero8();
    for (int kk = 0; kk < DIM / 4; ++kk) {
      int k0 = kk * 4 + lh * 2;
      v2f bf;
      bf.x = Wq[(size_t)k0 * DIM + n0 + lm];
      bf.y = Wq[(size_t)(k0 + 1) * DIM + n0 + lm];
      v2f af;
      af.x = sn[lm * DIM + k0];
      af.y = sn[lm * DIM + k0 + 1];
      acc = wmma4(af, bf, acc);
    }
    float bb = bq[n0 + lm];
#pragma unroll
    for (int v = 0; v < 8; ++v)
      qpad[((size_t)b * KPAD + v + lh * 8) * DIM + n0 + lm] = acc[v] + bb;
  }
}

// ---------------------------------------------------------------------------
// dots = q @ key^T * SCALE ; softmax over slot axis (11 valid of 16 rows);
// write attn (+EPS) to the output attn section.
// grid: (NTOK/128, BATCH), 256 threads (8 waves), one 16-col n-tile per wave.
// ---------------------------------------------------------------------------
__global__ __launch_bounds__(256) void dots_attn_kernel(
    const float* __restrict__ qpad, const float* __restrict__ key,
    float* __restrict__ attn) {
  __shared__ __align__(16) float qs[KPAD * DIM];
  const int b = blockIdx.y, tid = threadIdx.x;
#ifdef HAS_ASYNC_LDS
  for (int i = tid * 4; i < KPAD * DIM; i += 256 * 4)
    async_copy_b128(qpad + (size_t)b * KPAD * DIM + i, &qs[i]);
  wait_async0();
#else
  for (int i = tid; i < KPAD * DIM; i += 256) qs[i] = qpad[(size_t)b * KPAD * DIM + i];
#endif
  __syncthreads();

  const int wave = tid >> 5, lane = tid & 31;
  const int lm = lane & 15, lh = lane >> 4;
  const int n = blockIdx.x * 128 + wave * 16 + lm;
  const float* kb = key + (size_t)b * NTOK * DIM;

  v8f acc = vzero8();
  for (int kk = 0; kk < DIM / 4; ++kk) {
    int k0 = kk * 4 + lh * 2;
    v2f af; af.x = qs[lm * DIM + k0]; af.y = qs[lm * DIM + k0 + 1];
    v2f bf; bf.x = kb[(size_t)n * DIM + k0]; bf.y = kb[(size_t)n * DIM + k0 + 1];
    acc = wmma4(af, bf, acc);
  }

  float sc[8];
#pragma unroll
  for (int v = 0; v < 8; ++v) sc[v] = acc[v] * SCALE_;
  const int nval = lh ? 3 : 8;  // rows 0..7 in low half-lanes, 8..10 in high
  float mx = -3.0e38f;
#pragma unroll
  for (int v = 0; v < 8; ++v) if (v < nval) mx = fmaxf(mx, sc[v]);
  mx = fmaxf(mx, __shfl_xor(mx, 16, 32));
  float p[8]; float s = 0.f;
#pragma unroll
  for (int v = 0; v < 8; ++v) { p[v] = (v < nval) ? expf(sc[v] - mx) : 0.f; s += p[v]; }
  s += __shfl_xor(s, 16, 32);
  float invs = 1.f / s;
#pragma unroll
  for (int v = 0; v < 8; ++v)
    if (v < nval) {
      int k = v + lh * 8;
      attn[((size_t)(b * NSLOT + k)) * NTOK + n] = p[v] * invs + EPS_;
    }
}

// ---------------------------------------------------------------------------
// S[row] = sum_n attn[row][n]
// ---------------------------------------------------------------------------
__global__ __launch_bounds__(256) void rowsum_kernel(const float* __restrict__ attn,
                                                     float* __restrict__ S) {
  __shared__ float red[256];
  const int row = blockIdx.x, tid = threadIdx.x;
  float s = 0.f;
  for (int i = tid; i < NTOK; i += 256) s += attn[(size_t)row * NTOK + i];
  red[tid] = s;
  __syncthreads();
  for (int st = 128; st > 0; st >>= 1) {
    if (tid < st) red[tid] += red[tid + st];
    __syncthreads();
  }
  if (tid == 0) S[row] = red[0];
}

// ---------------------------------------------------------------------------
// updates = (attn/S) @ val : per batch [16(11)x4096]@[4096x192]
// grid: BATCH blocks, 384 threads (12 waves), one 16-col d-tile per wave.
// ---------------------------------------------------------------------------
__global__ __launch_bounds__(384) void updates_kernel(
    const float* __restrict__ attn, const float* __restrict__ S,
    const float* __restrict__ val, float* __restrict__ upd) {
  const int b = blockIdx.x, tid = threadIdx.x;
  const int wave = tid >> 5, lane = tid & 31;
  const int lm = lane & 15, lh = lane >> 4;
  const int d0 = wave * 16;
  const int mc = (lm < NSLOT) ? lm : (NSLOT - 1);
  const float invS = (lm < NSLOT) ? (1.f / S[b * NSLOT + mc]) : 0.f;
  const float* arow = attn + (size_t)(b * NSLOT + mc) * NTOK;
  const float* vb = val + (size_t)b * NTOK * DIM;

  v8f acc = vzero8();
#pragma unroll 4
  for (int kk = 0; kk < NTOK / 4; ++kk) {
    int k0 = kk * 4 + lh * 2;
    v2f af; af.x = arow[k0] * invS; af.y = arow[k0 + 1] * invS;
    v2f bf;
    bf.x = vb[(size_t)k0 * DIM + d0 + lm];
    bf.y = vb[(size_t)(k0 + 1) * DIM + d0 + lm];
    acc = wmma4(af, bf, acc);
  }
#pragma unroll
  for (int v = 0; v < 8; ++v) {
    int m = v + lh * 8;
    if (m < NSLOT)
      upd[(size_t)(b * NSLOT + m) * DIM + d0 + lm] = acc[v];
  }
}

// ---------------------------------------------------------------------------
// GRU cell per slot row (tiny: ~311 MFLOP per iter, ~934 MFLOP across 3 iters)
// ---------------------------------------------------------------------------
__global__ __launch_bounds__(192) void gru_kernel(
    const float* __restrict__ upd, const float* __restrict__ hprev,
    const float* __restrict__ Wih, const float* __restrict__ Whh,
    const float* __restrict__ bih, const float* __restrict__ bhh,
    float* __restrict__ hout) {
  __shared__ float xs[DIM], hs[DIM];
  const int row = blockIdx.x, t = threadIdx.x;
  xs[t] = upd[(size_t)row * DIM + t];
  hs[t] = hprev[(size_t)row * DIM + t];
  __syncthreads();
  float gx[3], gh[3];
  for (int g = 0; g < 3; ++g) {
    int j = g * DIM + t;
    const float* wi = Wih + (size_t)j * DIM;
    const float* wh = Whh + (size_t)j * DIM;
    float sx = bih[j], sh = bhh[j];
    for (int k = 0; k < DIM; ++k) { sx += xs[k] * wi[k]; sh += hs[k] * wh[k]; }
    gx[g] = sx; gh[g] = sh;
  }
  float r = 1.f / (1.f + expf(-(gx[0] + gh[0])));
  float z = 1.f / (1.f + expf(-(gx[1] + gh[1])));
  float nn = tanhf(gx[2] + r * gh[2]);
  hout[(size_t)row * DIM + t] = (1.f - z) * nn + z * hs[t];
}

// ---------------------------------------------------------------------------
// slots += MLP(LN(slots)) per row, in place
// ---------------------------------------------------------------------------
__global__ __launch_bounds__(192) void mlp_kernel(
    float* __restrict__ slots, const float* __restrict__ lnw,
    const float* __restrict__ lnb, const float* __restrict__ W1,
    const float* __restrict__ b1, const float* __restrict__ W2,
    const float* __restrict__ b2) {
  __shared__ float raw[DIM], mln[DIM], hid[HIDN], red[DIM];
  __shared__ float mu_s, inv_s;
  const int row = blockIdx.x, t = threadIdx.x;
  raw[t] = slots[(size_t)row * DIM + t];
  red[t] = raw[t];
  __syncthreads();
  if (t == 0) { float s = 0.f; for (int k = 0; k < DIM; ++k) s += red[k]; mu_s = s * (1.f / DIM); }
  __syncthreads();
  float d = raw[t] - mu_s;
  red[t] = d * d;
  __syncthreads();
  if (t == 0) { float s = 0.f; for (int k = 0; k < DIM; ++k) s += red[k]; inv_s = rsqrtf(s * (1.f / DIM) + LN_EPS_); }
  __syncthreads();
  mln[t] = d * inv_s * lnw[t] + lnb[t];
  __syncthreads();
  if (t < HIDN) {
    float s = b1[t];
    for (int k = 0; k < DIM; ++k) s += mln[k] * W1[(size_t)k * HIDN + t];
    hid[t] = fmaxf(s, 0.f);
  }
  __syncthreads();
  float o = b2[t];
  for (int h = 0; h < HIDN; ++h) o += hid[h] * W2[(size_t)h * DIM + t];
  slots[(size_t)row * DIM + t] = raw[t] + o;
}

__global__ void copy_kernel(const float* __restrict__ src, float* __restrict__ dst) {
  int i = blockIdx.x * 256 + threadIdx.x;
  if (i < BATCH * NSLOT * DIM) dst[i] = src[i];
}

// ---------------------------------------------------------------------------
extern "C" void kernel_launch(void* const* d_in, const int* in_sizes, int n_in,
                              void* d_out, int out_size, void* d_ws, size_t ws_size,
                              hipStream_t stream) {
  (void)in_sizes; (void)n_in; (void)out_size; (void)ws_size;
  const float* inputs      = (const float*)d_in[0];
  const float* noise       = (const float*)d_in[1];
  const float* slots_mu    = (const float*)d_in[2];
  const float* slots_sigma = (const float*)d_in[3];
  const float* ln_in_w     = (const float*)d_in[4];
  const float* ln_in_b     = (const float*)d_in[5];
  const float* ln_slots_w  = (const float*)d_in[6];
  const float* ln_slots_b  = (const float*)d_in[7];
  const float* ln_mlp_w    = (const float*)d_in[8];
  const float* ln_mlp_b    = (const float*)d_in[9];
  const float* Wq          = (const float*)d_in[10];
  const float* bq          = (const float*)d_in[11];
  const float* Wk          = (const float*)d_in[12];
  const float* bk          = (const float*)d_in[13];
  const float* Wv          = (const float*)d_in[14];
  const float* bv          = (const float*)d_in[15];
  const float* gru_Wih     = (const float*)d_in[16];
  const float* gru_Whh     = (const float*)d_in[17];
  const float* gru_bih     = (const float*)d_in[18];
  const float* gru_bhh     = (const float*)d_in[19];
  const float* mlp_W1      = (const float*)d_in[20];
  const float* mlp_b1      = (const float*)d_in[21];
  const float* mlp_W2      = (const float*)d_in[22];
  const float* mlp_b2      = (const float*)d_in[23];

  float* out_slots = (float*)d_out;
  float* out_attn  = out_slots + (size_t)BATCH * NSLOT * DIM;

  const size_t BND = (size_t)BATCH * NTOK * DIM;
  float* ws   = (float*)d_ws;
  float* key  = ws;
  float* val  = key + BND;
  float* qpad = val + BND;
  float* Ssum = qpad + (size_t)BATCH * KPAD * DIM;
  float* upd  = Ssum + 1024;
  float* sA   = upd + (size_t)BATCH * NSLOT * DIM;
  float* sB   = sA + (size_t)BATCH * NSLOT * DIM;

  ln_kv_kernel<<<(BATCH * NTOK) / 64, 256, 0, stream>>>(
      inputs, ln_in_w, ln_in_b, Wk, bk, Wv, bv, key, val);
  init_slots_kernel<<<(BATCH * NSLOT * DIM + 255) / 256, 256, 0, stream>>>(
      noise, slots_mu, slots_sigma, sA);

  float* cur = sA;
  float* nxt = sB;
  for (int it = 0; it < NITERS; ++it) {
    slots_q_kernel<<<BATCH, 128, 0, stream>>>(cur, ln_slots_w, ln_slots_b, Wq, bq, qpad);
    dots_attn_kernel<<<dim3(NTOK / 128, BATCH), 256, 0, stream>>>(qpad, key, out_attn);
    rowsum_kernel<<<BATCH * NSLOT, 256, 0, stream>>>(out_attn, Ssum);
    updates_kernel<<<BATCH, 384, 0, stream>>>(out_attn, Ssum, val, upd);
    gru_kernel<<<BATCH * NSLOT, 192, 0, stream>>>(
        upd, cur, gru_Wih, gru_Whh, gru_bih, gru_bhh, nxt);
    mlp_kernel<<<BATCH * NSLOT, 192, 0, stream>>>(
        nxt, ln_mlp_w, ln_mlp_b, mlp_W1, mlp_b1, mlp_W2, mlp_b2);
    float* tmp = cur; cur = nxt; nxt = tmp;
  }
  copy_kernel<<<(BATCH * NSLOT * DIM + 255) / 256, 256, 0, stream>>>(cur, out_slots);
}